// VQC_49589692400355
// MI455X (gfx1250) — compile-verified
//
#include <hip/hip_runtime.h>
#include <hip/hip_bf16.h>

#define NQ     12
#define NST    (1 << NQ)   // 4096 amplitudes
#define NLAYER 2
#define TPB    256         // 8 wave32s; each thread owns 16 amplitudes (4 local qubits)

// staging layout in LDS: x row [0..11], theta [12..59], lam [60..83]
#define STAGE_X     0
#define STAGE_TH    12
#define STAGE_LAM   60
#define STAGE_N     84

#if defined(__has_builtin)
#if __has_builtin(__builtin_amdgcn_global_load_async_to_lds_b128)
#define HAVE_ASYNC_LDS 1
#endif
#if __has_builtin(__builtin_amdgcn_s_wait_asynccnt)
#define HAVE_WAIT_ASYNC 1
#endif
#endif

typedef int v4i_ __attribute__((vector_size(16)));
typedef __attribute__((address_space(1))) v4i_ gv4i;   // global-AS int4
typedef __attribute__((address_space(3))) v4i_ lv4i;   // LDS-AS int4

// ---- compile-time index helpers -------------------------------------------
template<int L0, int L1, int L2, int L3>
__device__ __forceinline__ int off_of(int j) {
    return ((j & 1) << L0) | (((j >> 1) & 1) << L1) |
           (((j >> 2) & 1) << L2) | (((j >> 3) & 1) << L3);
}
template<int C0, int C1, int C2, int C3, int C4, int C5, int C6, int C7>
__device__ __forceinline__ int base_of(int tid) {
    return (((tid >> 0) & 1) << C0) | (((tid >> 1) & 1) << C1) |
           (((tid >> 2) & 1) << C2) | (((tid >> 3) & 1) << C3) |
           (((tid >> 4) & 1) << C4) | (((tid >> 5) & 1) << C5) |
           (((tid >> 6) & 1) << C6) | (((tid >> 7) & 1) << C7);
}

// ---- ownership remap through LDS: write old map, barrier, read new map.
// One barrier suffices: the next write sweep touches exactly the addresses this
// thread reads here (ownership is a partition), so there is no read/write race.
template<int L0, int L1, int L2, int L3,
         int A0, int A1, int A2, int A3, int A4, int A5, int A6, int A7,
         int M0, int M1, int M2, int M3,
         int B0, int B1, int B2, int B3, int B4, int B5, int B6, int B7>
__device__ __forceinline__ void remap(float* sre, float* sim_,
                                      float (&rr)[16], float (&ri)[16], int tid) {
    const int bo = base_of<A0, A1, A2, A3, A4, A5, A6, A7>(tid);
    #pragma unroll
    for (int j = 0; j < 16; ++j) {
        const int o = bo + off_of<L0, L1, L2, L3>(j);
        sre[o] = rr[j];
        sim_[o] = ri[j];
    }
    __syncthreads();
    const int bn = base_of<B0, B1, B2, B3, B4, B5, B6, B7>(tid);
    #pragma unroll
    for (int j = 0; j < 16; ++j) {
        const int o = bn + off_of<M0, M1, M2, M3>(j);
        rr[j] = sre[o];
        ri[j] = sim_[o];
    }
}

// ---- register-resident gates ----------------------------------------------
template<int K>
__device__ __forceinline__ void rx_gate(float (&rr)[16], float (&ri)[16], float c, float s) {
    #pragma unroll
    for (int j = 0; j < 16; ++j) {
        if (j & (1 << K)) continue;
        const int j1 = j | (1 << K);
        float a0r = rr[j], a0i = ri[j], a1r = rr[j1], a1i = ri[j1];
        rr[j]  = c * a0r + s * a1i;
        ri[j]  = c * a0i - s * a1r;
        rr[j1] = c * a1r + s * a0i;
        ri[j1] = c * a1i - s * a0r;
    }
}
template<int KC, int KT>
__device__ __forceinline__ void cnot_gate(float (&rr)[16], float (&ri)[16]) {
    #pragma unroll
    for (int j = 0; j < 16; ++j) {
        if (!(j & (1 << KC)) || (j & (1 << KT))) continue;
        const int j1 = j | (1 << KT);
        float tr = rr[j], ti = ri[j];
        rr[j] = rr[j1]; ri[j] = ri[j1];
        rr[j1] = tr;    ri[j1] = ti;
    }
}
template<int K>
__device__ __forceinline__ void ryrz_gate(float (&rr)[16], float (&ri)[16],
                                          float cy, float sy, float cz, float sz) {
    #pragma unroll
    for (int j = 0; j < 16; ++j) {
        if (j & (1 << K)) continue;
        const int j1 = j | (1 << K);
        float a0r = rr[j], a0i = ri[j], a1r = rr[j1], a1i = ri[j1];
        float n0r = cy * a0r - sy * a1r, n0i = cy * a0i - sy * a1i;
        float n1r = sy * a0r + cy * a1r, n1i = sy * a0i + cy * a1i;
        rr[j]  = n0r * cz + n0i * sz;   // * e^{-i hz}
        ri[j]  = n0i * cz - n0r * sz;
        rr[j1] = n1r * cz - n1i * sz;   // * e^{+i hz}
        ri[j1] = n1i * cz + n1r * sz;
    }
}

__device__ __forceinline__ float ldc(const float* coef, int l, int q, int k) {
    return coef[(l * NQ + q) * 6 + k];
}

__global__ __launch_bounds__(TPB) void vqc_statevec_kernel(
    const float* __restrict__ x,      // [B, NQ]
    const float* __restrict__ theta,  // [NLAYER, NQ, 2]
    const float* __restrict__ lam,    // [NLAYER, NQ]
    float* __restrict__ out)          // [B, NQ]
{
    __shared__ float sre[NST];               // 16 KB
    __shared__ float sim_[NST];              // 16 KB
    __shared__ float coef[NLAYER * NQ * 6];
    __shared__ float zacc[NQ];
    __shared__ __align__(16) float stage[STAGE_N];

    const int tid = threadIdx.x;
    const int b   = blockIdx.x;

    if (tid == 0) {   // gfx1250 global_prefetch_b8
        __builtin_prefetch(x + b * NQ, 0, 3);
        __builtin_prefetch(theta, 0, 3);
        __builtin_prefetch(lam, 0, 3);
    }
    if (tid < NQ) zacc[tid] = 0.0f;

#if defined(HAVE_ASYNC_LDS)
    // ---- async DMA of parameters into LDS staging (wave 0, 21 lanes x 16B) ----
    if (tid < 21) {
        const float* g;
        float* lp;
        if (tid < 3)       { g = x + b * NQ + tid * 4;   lp = &stage[STAGE_X  + tid * 4]; }
        else if (tid < 15) { g = theta + (tid - 3) * 4;  lp = &stage[STAGE_TH + (tid - 3) * 4]; }
        else               { g = lam + (tid - 15) * 4;   lp = &stage[STAGE_LAM + (tid - 15) * 4]; }
        __builtin_amdgcn_global_load_async_to_lds_b128(
            (gv4i*)(v4i_*)(void*)g,   // addrspacecast generic -> AS1, pointee int4
            (lv4i*)(v4i_*)(void*)lp,  // addrspacecast generic -> AS3, pointee int4
            0, 0);
    }
#if defined(HAVE_WAIT_ASYNC)
    __builtin_amdgcn_s_wait_asynccnt(0);
#else
    asm volatile("s_wait_asynccnt 0x0" ::: "memory");
#endif
#else
    // fallback: plain cooperative copy into staging
    if (tid < STAGE_N) {
        stage[tid] = (tid < STAGE_TH)  ? x[b * NQ + tid]
                   : (tid < STAGE_LAM) ? theta[tid - STAGE_TH]
                                       : lam[tid - STAGE_LAM];
    }
#endif
    __syncthreads();

    // per-gate trig coefficients (lanes 0..23 cover (l,q)), sourced from LDS stage
    if (tid < NLAYER * NQ) {
        const int l = tid / NQ, q = tid % NQ;
        float s, c;
        sincosf(0.5f * stage[STAGE_LAM + l * NQ + q] * stage[STAGE_X + q], &s, &c);
        coef[tid * 6 + 0] = c; coef[tid * 6 + 1] = s;
        sincosf(0.5f * stage[STAGE_TH + (l * NQ + q) * 2 + 0], &s, &c);
        coef[tid * 6 + 2] = c; coef[tid * 6 + 3] = s;
        sincosf(0.5f * stage[STAGE_TH + (l * NQ + q) * 2 + 1], &s, &c);
        coef[tid * 6 + 4] = c; coef[tid * 6 + 5] = s;
    }
    __syncthreads();

    // state lives in registers: thread owns 16 amps; start in P1 coords {0,1,2,3}
    float rr[16], ri[16];
    #pragma unroll
    for (int j = 0; j < 16; ++j) { rr[j] = 0.0f; ri[j] = 0.0f; }
    if (tid == 0) rr[0] = 1.0f;   // |0...0>

    #pragma unroll
    for (int l = 0; l < NLAYER; ++l) {
        // ---- P1: locals {0,1,2,3} ----
        rx_gate<0>(rr, ri, ldc(coef, l, 0, 0), ldc(coef, l, 0, 1));
        rx_gate<1>(rr, ri, ldc(coef, l, 1, 0), ldc(coef, l, 1, 1));
        rx_gate<2>(rr, ri, ldc(coef, l, 2, 0), ldc(coef, l, 2, 1));
        rx_gate<3>(rr, ri, ldc(coef, l, 3, 0), ldc(coef, l, 3, 1));
        cnot_gate<0, 1>(rr, ri);            // CNOT(0,1)
        cnot_gate<1, 2>(rr, ri);            // CNOT(1,2)
        cnot_gate<2, 3>(rr, ri);            // CNOT(2,3)
        remap<0,1,2,3,  4,5,6,7,8,9,10,11,
              3,4,5,6,  0,1,2,7,8,9,10,11>(sre, sim_, rr, ri, tid);

        // ---- P2: locals {3,4,5,6} ----
        rx_gate<1>(rr, ri, ldc(coef, l, 4, 0), ldc(coef, l, 4, 1));
        rx_gate<2>(rr, ri, ldc(coef, l, 5, 0), ldc(coef, l, 5, 1));
        rx_gate<3>(rr, ri, ldc(coef, l, 6, 0), ldc(coef, l, 6, 1));
        cnot_gate<0, 1>(rr, ri);            // CNOT(3,4)
        cnot_gate<1, 2>(rr, ri);            // CNOT(4,5)
        cnot_gate<2, 3>(rr, ri);            // CNOT(5,6)
        remap<3,4,5,6,  0,1,2,7,8,9,10,11,
              6,7,8,9,  0,1,2,3,4,5,10,11>(sre, sim_, rr, ri, tid);

        // ---- P3: locals {6,7,8,9} ----
        rx_gate<1>(rr, ri, ldc(coef, l, 7, 0), ldc(coef, l, 7, 1));
        rx_gate<2>(rr, ri, ldc(coef, l, 8, 0), ldc(coef, l, 8, 1));
        rx_gate<3>(rr, ri, ldc(coef, l, 9, 0), ldc(coef, l, 9, 1));
        cnot_gate<0, 1>(rr, ri);            // CNOT(6,7)
        cnot_gate<1, 2>(rr, ri);            // CNOT(7,8)
        cnot_gate<2, 3>(rr, ri);            // CNOT(8,9)
        remap<6,7,8,9,  0,1,2,3,4,5,10,11,
              9,10,11,0, 1,2,3,4,5,6,7,8>(sre, sim_, rr, ri, tid);

        // ---- P4: locals {9,10,11,0} (k0=9,k1=10,k2=11,k3=0) ----
        rx_gate<1>(rr, ri, ldc(coef, l, 10, 0), ldc(coef, l, 10, 1));
        rx_gate<2>(rr, ri, ldc(coef, l, 11, 0), ldc(coef, l, 11, 1));
        cnot_gate<0, 1>(rr, ri);            // CNOT(9,10)
        cnot_gate<1, 2>(rr, ri);            // CNOT(10,11)
        cnot_gate<2, 3>(rr, ri);            // CNOT(11,0)
        ryrz_gate<0>(rr, ri, ldc(coef, l,  9, 2), ldc(coef, l,  9, 3), ldc(coef, l,  9, 4), ldc(coef, l,  9, 5));
        ryrz_gate<1>(rr, ri, ldc(coef, l, 10, 2), ldc(coef, l, 10, 3), ldc(coef, l, 10, 4), ldc(coef, l, 10, 5));
        ryrz_gate<2>(rr, ri, ldc(coef, l, 11, 2), ldc(coef, l, 11, 3), ldc(coef, l, 11, 4), ldc(coef, l, 11, 5));
        ryrz_gate<3>(rr, ri, ldc(coef, l,  0, 2), ldc(coef, l,  0, 3), ldc(coef, l,  0, 4), ldc(coef, l,  0, 5));
        remap<9,10,11,0, 1,2,3,4,5,6,7,8,
              1,2,3,4,   0,5,6,7,8,9,10,11>(sre, sim_, rr, ri, tid);

        // ---- P5: locals {1,2,3,4} ----
        ryrz_gate<0>(rr, ri, ldc(coef, l, 1, 2), ldc(coef, l, 1, 3), ldc(coef, l, 1, 4), ldc(coef, l, 1, 5));
        ryrz_gate<1>(rr, ri, ldc(coef, l, 2, 2), ldc(coef, l, 2, 3), ldc(coef, l, 2, 4), ldc(coef, l, 2, 5));
        ryrz_gate<2>(rr, ri, ldc(coef, l, 3, 2), ldc(coef, l, 3, 3), ldc(coef, l, 3, 4), ldc(coef, l, 3, 5));
        ryrz_gate<3>(rr, ri, ldc(coef, l, 4, 2), ldc(coef, l, 4, 3), ldc(coef, l, 4, 4), ldc(coef, l, 4, 5));
        remap<1,2,3,4,  0,5,6,7,8,9,10,11,
              5,6,7,8,  0,1,2,3,4,9,10,11>(sre, sim_, rr, ri, tid);

        // ---- P6: locals {5,6,7,8} ----
        ryrz_gate<0>(rr, ri, ldc(coef, l, 5, 2), ldc(coef, l, 5, 3), ldc(coef, l, 5, 4), ldc(coef, l, 5, 5));
        ryrz_gate<1>(rr, ri, ldc(coef, l, 6, 2), ldc(coef, l, 6, 3), ldc(coef, l, 6, 4), ldc(coef, l, 6, 5));
        ryrz_gate<2>(rr, ri, ldc(coef, l, 7, 2), ldc(coef, l, 7, 3), ldc(coef, l, 7, 4), ldc(coef, l, 7, 5));
        ryrz_gate<3>(rr, ri, ldc(coef, l, 8, 2), ldc(coef, l, 8, 3), ldc(coef, l, 8, 4), ldc(coef, l, 8, 5));

        if (l + 1 < NLAYER) {   // back to P1 coords for the next layer
            remap<5,6,7,8,  0,1,2,3,4,9,10,11,
                  0,1,2,3,  4,5,6,7,8,9,10,11>(sre, sim_, rr, ri, tid);
        }
    }

    // ---- measurement straight from registers (P6 coords: locals {5,6,7,8}) ----
    float zl0 = 0.f, zl1 = 0.f, zl2 = 0.f, zl3 = 0.f, ptot = 0.f;
    #pragma unroll
    for (int j = 0; j < 16; ++j) {
        float p = rr[j] * rr[j] + ri[j] * ri[j];
        ptot += p;
        zl0 += (j & 1) ? -p : p;   // qubit 5
        zl1 += (j & 2) ? -p : p;   // qubit 6
        zl2 += (j & 4) ? -p : p;   // qubit 7
        zl3 += (j & 8) ? -p : p;   // qubit 8
    }
    const int base = base_of<0, 1, 2, 3, 4, 9, 10, 11>(tid);
    float z[NQ];
    z[5] = zl0; z[6] = zl1; z[7] = zl2; z[8] = zl3;
    #pragma unroll
    for (int q = 0; q < 5; ++q)  z[q] = (base & (1 << q)) ? -ptot : ptot;
    #pragma unroll
    for (int q = 9; q < NQ; ++q) z[q] = (base & (1 << q)) ? -ptot : ptot;

    // wave32 shuffle reduction, then one ds_add_f32 per wave per qubit
    #pragma unroll
    for (int q = 0; q < NQ; ++q) {
        #pragma unroll
        for (int off = 16; off > 0; off >>= 1)
            z[q] += __shfl_down(z[q], off, 32);
    }
    if ((tid & 31) == 0) {
        #pragma unroll
        for (int q = 0; q < NQ; ++q)
            atomicAdd(&zacc[q], z[q]);
    }
    __syncthreads();

    if (tid < NQ) out[b * NQ + tid] = zacc[tid];
}

extern "C" void kernel_launch(void* const* d_in, const int* in_sizes, int n_in,
                              void* d_out, int out_size, void* d_ws, size_t ws_size,
                              hipStream_t stream) {
    const float* x     = (const float*)d_in[0];  // [1024, 12]
    const float* theta = (const float*)d_in[1];  // [2, 12, 2]
    const float* lam   = (const float*)d_in[2];  // [2, 12]
    float* out = (float*)d_out;                  // [1024, 12]

    const int batch = in_sizes[0] / NQ;          // 1024
    vqc_statevec_kernel<<<batch, TPB, 0, stream>>>(x, theta, lam, out);
}